// AdvancedGraphNeuralNetwork_56349970924160
// MI455X (gfx1250) — compile-verified
//
#include <hip/hip_runtime.h>
#include <hip/hip_bf16.h>
#include <math.h>

// ---------------------------------------------------------------------------
// Problem constants (match reference setup_inputs)
// ---------------------------------------------------------------------------
#define N_NODES 20000
#define N_EDGES 320000
#define D0 256
#define D1 256
#define D2 128
#define D3 64
#define GN_EPS 1e-5f

typedef __attribute__((ext_vector_type(2))) float v2f;
typedef __attribute__((ext_vector_type(8))) float v8f;

// ---------------------------------------------------------------------------
// Generic fill
// ---------------------------------------------------------------------------
__global__ void fill_f32(float* __restrict__ p, float v, int count) {
    int i = blockIdx.x * blockDim.x + threadIdx.x;
    if (i < count) p[i] = v;
}

// ---------------------------------------------------------------------------
// Degree: deg initialized to 1.0 (self loop), count incoming edges, then
// deg[i] <- rsqrt(deg[i])  (deg >= 1 always)
// ---------------------------------------------------------------------------
__global__ void deg_count(const int* __restrict__ dst, float* __restrict__ deg, int E) {
    int i = blockIdx.x * blockDim.x + threadIdx.x;
    if (i < E) atomicAdd(&deg[dst[i]], 1.0f);
}

__global__ void deg_to_dinv(float* __restrict__ deg, int n) {
    int i = blockIdx.x * blockDim.x + threadIdx.x;
    if (i < n) deg[i] = rsqrtf(deg[i]);
}

// ---------------------------------------------------------------------------
// Y[n,fout] = X[n,fin] @ W[fin,fout], fp32 via V_WMMA_F32_16X16X4_F32.
// One wave computes one 16x16 tile of Y; K advances 4 per WMMA.
// A (16x4 f32): lanes 0-15 M=lane, lanes 16-31 M=lane-16; VGPR0/1 = K pair,
//   lane-half selects K base (0 or 2).
// B (4x16 f32): symmetric — lanes hold N=lane&15, K = 2*(lane>>4)+vgpr.
// C/D (16x16 f32): VGPR r -> M=r (lanes 0-15) / M=r+8 (lanes 16-31), N=lane&15.
// Tile index is wave-uniform so the early-out keeps EXEC all-ones for WMMA.
// ---------------------------------------------------------------------------
__global__ void gemm_wmma_f32(const float* __restrict__ X,
                              const float* __restrict__ W,
                              float* __restrict__ Y,
                              int n, int fin, int fout) {
    const int lane = threadIdx.x & 31;
    const int wave = threadIdx.x >> 5;
    const int tilesN = fout >> 4;
    const int numTiles = (n >> 4) * tilesN;
    int tile = blockIdx.x * (blockDim.x >> 5) + wave;
    if (tile >= numTiles) return;

    const int tm = tile / tilesN;
    const int tn = tile - tm * tilesN;
    const int row0 = tm << 4;
    const int col0 = tn << 4;
    const int half = lane >> 4;   // 0 or 1
    const int l16  = lane & 15;

    const float* __restrict__ xr = X + (size_t)(row0 + l16) * fin;  // A row M=l16
    const float* __restrict__ wc = W + col0 + l16;                  // B col N=l16

    v8f c = {};
    for (int k = 0; k < fin; k += 4) {
        const int ka = k + (half << 1);
        v2f a, b;
        a.x = xr[ka];
        a.y = xr[ka + 1];
        b.x = wc[(size_t)ka * fout];
        b.y = wc[(size_t)(ka + 1) * fout];
        c = __builtin_amdgcn_wmma_f32_16x16x4_f32(
                /*neg_a=*/false, a, /*neg_b=*/false, b,
                /*c_mod=*/(short)0, c, /*reuse_a=*/false, /*reuse_b=*/false);
    }

#pragma unroll
    for (int r = 0; r < 8; ++r)
        Y[(size_t)(row0 + r + 8 * half) * fout + col0 + l16] = c[r];
}

// ---------------------------------------------------------------------------
// out[i,f] = y[i,f]*dinv[i]^2 + bias[f]     (self-loop message + bias)
// fout is a power of two -> foutShift
// ---------------------------------------------------------------------------
__global__ void init_out(const float* __restrict__ y,
                         const float* __restrict__ dinv,
                         const float* __restrict__ bias,
                         float* __restrict__ out,
                         int total, int foutShift) {
    int idx = blockIdx.x * blockDim.x + threadIdx.x;
    if (idx >= total) return;
    const int i = idx >> foutShift;
    const int f = idx & ((1 << foutShift) - 1);
    const float di = dinv[i];
    out[idx] = y[idx] * di * di + bias[f];
}

// ---------------------------------------------------------------------------
// Edge scatter: out[dst,f] += y[src,f] * dinv[src]*dinv[dst]
// blockDim = 256; (256 >> foutShift) edges per block; lanes within an edge
// cover consecutive features -> fully coalesced gather of y[src,:].
// Random accesses land in L2 (y and out both fit in the 192 MB L2).
// ---------------------------------------------------------------------------
__global__ void edge_scatter(const float* __restrict__ y,
                             const int* __restrict__ src,
                             const int* __restrict__ dst,
                             const float* __restrict__ dinv,
                             float* __restrict__ out,
                             int E, int foutShift) {
    const int fout = 1 << foutShift;
    const int epb  = blockDim.x >> foutShift;
    const int e = blockIdx.x * epb + (threadIdx.x >> foutShift);
    if (e >= E) return;
    const int f = threadIdx.x & (fout - 1);
    const int s = src[e];
    const int d = dst[e];
    const float nrm = dinv[s] * dinv[d];
    atomicAdd(&out[(size_t)d * fout + f], y[(size_t)s * fout + f] * nrm);
}

// ---------------------------------------------------------------------------
// Column reduction: per-feature sum and sum-of-squares over n rows.
// blockDim = fout (64/128/256). Each block handles a contiguous row slab,
// one partial per feature, then a single atomicAdd per feature.
// ---------------------------------------------------------------------------
__global__ void col_reduce(const float* __restrict__ x,
                           float* __restrict__ csum,
                           float* __restrict__ csumsq,
                           int n, int rowsPerBlock) {
    const int fout = blockDim.x;
    const int f = threadIdx.x;
    int r0 = blockIdx.x * rowsPerBlock;
    int r1 = r0 + rowsPerBlock;
    if (r1 > n) r1 = n;
    float s = 0.0f, s2 = 0.0f;
    for (int r = r0; r < r1; ++r) {
        const float v = x[(size_t)r * fout + f];
        s += v;
        s2 += v * v;
    }
    atomicAdd(&csum[f], s);
    atomicAdd(&csumsq[f], s2);
}

// ---------------------------------------------------------------------------
// GraphNorm + exact GELU + optional residual, in place.
// mean m = csum/n ; var = E[x^2] - 2a m^2 + a^2 m^2  (== E[(x-am)^2])
// ---------------------------------------------------------------------------
__global__ void apply_norm_gelu(const float* __restrict__ xin,
                                const float* __restrict__ csum,
                                const float* __restrict__ csumsq,
                                const float* __restrict__ ga,
                                const float* __restrict__ gw,
                                const float* __restrict__ gb,
                                const float* __restrict__ xres,  // may be null
                                float* __restrict__ xout,
                                int total, int foutShift, float invn) {
    int idx = blockIdx.x * blockDim.x + threadIdx.x;
    if (idx >= total) return;
    const int f = idx & ((1 << foutShift) - 1);
    const float m = csum[f] * invn;
    const float a = ga[f];
    const float var = csumsq[f] * invn - 2.0f * a * m * m + a * a * m * m;
    const float v = xin[idx];
    const float t = gw[f] * (v - a * m) * rsqrtf(var + GN_EPS) + gb[f];
    float g = 0.5f * t * (1.0f + erff(t * 0.70710678118654752f));
    if (xres) g += xres[idx];
    xout[idx] = g;
}

// ---------------------------------------------------------------------------
// CAS float max (ordered for non-NaN values)
// ---------------------------------------------------------------------------
__device__ inline void atomicMaxF32(float* addr, float val) {
    unsigned int* ua = (unsigned int*)addr;
    unsigned int old = *ua;
    while (__uint_as_float(old) < val) {
        unsigned int assumed = old;
        old = atomicCAS(ua, assumed, __float_as_uint(val));
        if (old == assumed) break;
    }
}

// ---------------------------------------------------------------------------
// Pool: per-feature sum and max over n rows of x [n,64]. blockDim = 64.
// ---------------------------------------------------------------------------
__global__ void pool_reduce(const float* __restrict__ x,
                            float* __restrict__ psum,
                            float* __restrict__ pmax,
                            int n, int rowsPerBlock) {
    const int f = threadIdx.x;   // 0..63
    int r0 = blockIdx.x * rowsPerBlock;
    int r1 = r0 + rowsPerBlock;
    if (r1 > n) r1 = n;
    float s = 0.0f, mx = -INFINITY;
    for (int r = r0; r < r1; ++r) {
        const float v = x[(size_t)r * D3 + f];
        s += v;
        mx = fmaxf(mx, v);
    }
    atomicAdd(&psum[f], s);
    atomicMaxF32(&pmax[f], mx);
}

// ---------------------------------------------------------------------------
// Head: pooled = [mean | max | sum] (192) -> wc1 (192x64) + bc1 -> LayerNorm
// -> GELU -> wc2 (64x2) + bc2. One block, 64 threads.
// ---------------------------------------------------------------------------
__global__ void head_kernel(const float* __restrict__ psum,
                            const float* __restrict__ pmax,
                            const float* __restrict__ wc1,
                            const float* __restrict__ bc1,
                            const float* __restrict__ lnw,
                            const float* __restrict__ lnb,
                            const float* __restrict__ wc2,
                            const float* __restrict__ bc2,
                            float* __restrict__ out,
                            float invn) {
    __shared__ float p[3 * D3];
    __shared__ float h[D3];
    __shared__ float red[D3];
    const int j = threadIdx.x;   // 0..63

    const float s = psum[j];
    p[j]          = s * invn;    // mean
    p[D3 + j]     = pmax[j];     // max
    p[2 * D3 + j] = s;           // sum
    __syncthreads();

    float acc = bc1[j];
    for (int k = 0; k < 3 * D3; ++k) acc += p[k] * wc1[(size_t)k * D3 + j];

    // LayerNorm over the 64 features
    red[j] = acc;
    __syncthreads();
    for (int st = 32; st > 0; st >>= 1) {
        if (j < st) red[j] += red[j + st];
        __syncthreads();
    }
    const float mean = red[0] * (1.0f / D3);
    __syncthreads();
    const float cdiff = acc - mean;
    red[j] = cdiff * cdiff;
    __syncthreads();
    for (int st = 32; st > 0; st >>= 1) {
        if (j < st) red[j] += red[j + st];
        __syncthreads();
    }
    const float var = red[0] * (1.0f / D3);
    const float t = cdiff * rsqrtf(var + GN_EPS) * lnw[j] + lnb[j];
    h[j] = 0.5f * t * (1.0f + erff(t * 0.70710678118654752f));
    __syncthreads();

    if (j < 2) {
        float o = bc2[j];
        for (int k = 0; k < D3; ++k) o += h[k] * wc2[k * 2 + j];
        out[j] = o;
    }
}

// ---------------------------------------------------------------------------
// Host-side orchestration
// ---------------------------------------------------------------------------
static inline int ceil_div(int a, int b) { return (a + b - 1) / b; }

static void run_layer(const float* xin, const float* w, const float* b,
                      const float* gw, const float* gb, const float* ga,
                      const float* xres, float* y, float* xout,
                      float* csum, float* csumsq,
                      const int* src, const int* dst, const float* dinv,
                      int fin, int fout, int foutShift, hipStream_t stream) {
    const int n = N_NODES;
    const int total = n * fout;

    // 1. GEMM: y = xin @ w
    {
        const int numTiles = (n >> 4) * (fout >> 4);
        const int wavesPerBlock = 8;  // 256 threads
        gemm_wmma_f32<<<ceil_div(numTiles, wavesPerBlock), 256, 0, stream>>>(
            xin, w, y, n, fin, fout);
    }
    // 2. out = self-loop message + bias
    init_out<<<ceil_div(total, 256), 256, 0, stream>>>(y, dinv, b, xout, total, foutShift);
    // 3. edge scatter (atomic)
    {
        const int epb = 256 >> foutShift;
        edge_scatter<<<ceil_div(N_EDGES, epb), 256, 0, stream>>>(
            y, src, dst, dinv, xout, N_EDGES, foutShift);
    }
    // 4. per-feature sum / sumsq
    fill_f32<<<ceil_div(2 * fout, 256), 256, 0, stream>>>(csum, 0.0f, 2 * fout);
    {
        const int rowsPerBlock = 100;
        col_reduce<<<ceil_div(n, rowsPerBlock), fout, 0, stream>>>(
            xout, csum, csumsq, n, rowsPerBlock);
    }
    // 5. GraphNorm + GELU (+ residual), in place
    apply_norm_gelu<<<ceil_div(total, 256), 256, 0, stream>>>(
        xout, csum, csumsq, ga, gw, gb, xres, xout, total, foutShift, 1.0f / n);
}

extern "C" void kernel_launch(void* const* d_in, const int* in_sizes, int n_in,
                              void* d_out, int out_size, void* d_ws, size_t ws_size,
                              hipStream_t stream) {
    (void)in_sizes; (void)n_in; (void)out_size; (void)ws_size;

    const float* x     = (const float*)d_in[0];
    const int*   ei    = (const int*)d_in[1];
    const float* w0    = (const float*)d_in[2];
    const float* b0    = (const float*)d_in[3];
    const float* gn0w  = (const float*)d_in[4];
    const float* gn0b  = (const float*)d_in[5];
    const float* gn0a  = (const float*)d_in[6];
    const float* w1    = (const float*)d_in[7];
    const float* b1    = (const float*)d_in[8];
    const float* gn1w  = (const float*)d_in[9];
    const float* gn1b  = (const float*)d_in[10];
    const float* gn1a  = (const float*)d_in[11];
    const float* w2    = (const float*)d_in[12];
    const float* b2    = (const float*)d_in[13];
    const float* gn2w  = (const float*)d_in[14];
    const float* gn2b  = (const float*)d_in[15];
    const float* gn2a  = (const float*)d_in[16];
    const float* wc1   = (const float*)d_in[17];
    const float* bc1   = (const float*)d_in[18];
    const float* lnw   = (const float*)d_in[19];
    const float* lnb   = (const float*)d_in[20];
    const float* wc2   = (const float*)d_in[21];
    const float* bc2   = (const float*)d_in[22];

    const int* src = ei;             // edge_index[0]
    const int* dst = ei + N_EDGES;   // edge_index[1]

    // Workspace carve-up (floats)
    float* ws   = (float*)d_ws;
    float* y    = ws;                           // max 20000*256
    float* x1   = y    + (size_t)N_NODES * D1;  // 20000*256
    float* x2   = x1   + (size_t)N_NODES * D1;  // 20000*128
    float* x3   = x2   + (size_t)N_NODES * D2;  // 20000*64
    float* dinv = x3   + (size_t)N_NODES * D3;  // 20000
    float* csum = dinv + N_NODES;               // 256 (sum) + 256 (sumsq)
    float* csumsq = csum + D1;
    float* psum = csumsq + D1;                  // 64
    float* pmax = psum + D3;                    // 64

    // --- degrees / dinv (self loops included: deg starts at 1) ---
    fill_f32<<<ceil_div(N_NODES, 256), 256, 0, stream>>>(dinv, 1.0f, N_NODES);
    deg_count<<<ceil_div(N_EDGES, 256), 256, 0, stream>>>(dst, dinv, N_EDGES);
    deg_to_dinv<<<ceil_div(N_NODES, 256), 256, 0, stream>>>(dinv, N_NODES);

    // --- layer 0: 256 -> 256, residual ---
    run_layer(x,  w0, b0, gn0w, gn0b, gn0a, /*xres=*/x, y, x1,
              csum, csumsq, src, dst, dinv, D0, D1, 8, stream);
    // --- layer 1: 256 -> 128 ---
    run_layer(x1, w1, b1, gn1w, gn1b, gn1a, /*xres=*/nullptr, y, x2,
              csum, csumsq, src, dst, dinv, D1, D2, 7, stream);
    // --- layer 2: 128 -> 64 ---
    run_layer(x2, w2, b2, gn2w, gn2b, gn2a, /*xres=*/nullptr, y, x3,
              csum, csumsq, src, dst, dinv, D2, D3, 6, stream);

    // --- pooling: mean | max | sum over nodes ---
    fill_f32<<<1, 64, 0, stream>>>(psum, 0.0f, D3);
    fill_f32<<<1, 64, 0, stream>>>(pmax, -INFINITY, D3);
    pool_reduce<<<ceil_div(N_NODES, 100), 64, 0, stream>>>(x3, psum, pmax, N_NODES, 100);

    // --- head ---
    head_kernel<<<1, 64, 0, stream>>>(psum, pmax, wc1, bc1, lnw, lnb, wc2, bc2,
                                      (float*)d_out, 1.0f / N_NODES);
}